// SS2D_54966991454773
// MI455X (gfx1250) — compile-verified
//
#include <hip/hip_runtime.h>
#include <hip/hip_bf16.h>
#include <math.h>

// Problem constants (from reference setup_inputs)
#define BB 2
#define CC 192
#define TT 4
#define HH 32
#define WW 32
#define DP 384
#define NN 16
#define RR 12
#define KK 4
#define LL 1024           // H*W
#define BT 8              // B*T
#define RN 44             // R + 2N

typedef float v2f __attribute__((ext_vector_type(2)));
typedef float v8f __attribute__((ext_vector_type(8)));
typedef unsigned int v4u __attribute__((ext_vector_type(4)));
typedef unsigned int v8u __attribute__((ext_vector_type(8)));

// ---------------------------------------------------------------------------
// Pack x (B,C,T,H,W) -> xf [bt][C][L]  (bt = b*T+t)
// ---------------------------------------------------------------------------
__global__ void ss2d_pack_xf(const float* __restrict__ x, float* __restrict__ xf) {
  int i = blockIdx.x * blockDim.x + threadIdx.x;
  if (i >= BT * CC * LL) return;
  int l  = i % LL;
  int c  = (i / LL) % CC;
  int bt = i / (LL * CC);
  int b = bt / TT, t = bt % TT;
  xf[i] = x[(((size_t)(b * CC + c) * TT + t) * LL) + l];
}

// ---------------------------------------------------------------------------
// Generic fp32 WMMA GEMM:  Y[batch][m][l] = act( sum_k W[m,k] * X[batch][k][l] )
//   W     : row-major [M x Kd], row stride Kd
//   X     : batched   [Ntot/LL][KS][LL]   (only rows 0..Kd-1 used)
//   Y     : batched   [Ntot/LL][M ][LL]
//   act=1 : softplus(v + bias[m])   (for delta)
//
// One wave32 per 16x16 tile. The 16xKd A-tile is staged into LDS by the
// Tensor Data Mover (2D descriptor; rows past M are zero-filled by TDM OOB
// semantics, so no divergent guards are needed). B is double-buffered from
// global memory; accumulation runs on V_WMMA_F32_16X16X4_F32.
// ---------------------------------------------------------------------------
__global__ void ss2d_gemm_wmma_f32(const float* __restrict__ Wm,
                                   const float* __restrict__ Xm,
                                   float* __restrict__ Ym,
                                   const float* __restrict__ bias,
                                   int M, int Kd, int KS, int act) {
  __shared__ float As[16 * 384];   // max Kd = 384 -> 24 KB

  const int lane = threadIdx.x & 31;
  const int lm   = lane & 15;     // column within tile / A row
  const int half = lane >> 4;     // K-phase selector
  const int n0   = blockIdx.x * 16;
  const int m0   = blockIdx.y * 16;
  const int batch = n0 / LL;
  const int nloc  = (n0 % LL) + lm;

  // ---- TDM: stage A tile (16 rows x Kd elems of W, starting at row m0) ----
  {
    const unsigned lds  =
        (unsigned)__builtin_amdgcn_readfirstlane((int)(size_t)&As[0]);
    const unsigned long long ga =
        (unsigned long long)(size_t)(Wm + (size_t)m0 * Kd);
    const unsigned galo = (unsigned)ga;
    const unsigned gahi = (unsigned)(ga >> 32);
    const unsigned kd   = (unsigned)Kd;
    const unsigned dim1 = (unsigned)(M - m0);   // valid rows; rest zero-fill

    v4u g0;
    g0[0] = 1u;                                    // count=1, user mode
    g0[1] = lds;                                   // lds_addr
    g0[2] = galo;                                  // global_addr[31:0]
    g0[3] = (gahi & 0x01FFFFFFu) | 0x80000000u;    // addr[56:32] | type=2

    v8u g1;
    g1[0] = 0x00020000u;                           // data_size = 4B
    g1[1] = (kd & 0xFFFFu) << 16;                  // tensor_dim0 lo16
    g1[2] = (kd >> 16) | ((dim1 & 0xFFFFu) << 16); // dim0 hi | tensor_dim1 lo
    g1[3] = (dim1 >> 16) | ((kd & 0xFFFFu) << 16); // dim1 hi | tile_dim0=Kd
    g1[4] = 16u;                                   // tile_dim1 = 16 rows
    g1[5] = kd;                                    // tensor_dim0_stride lo32
    g1[6] = 0u;
    g1[7] = 0u;

    asm volatile("tensor_load_to_lds %0, %1" :: "s"(g0), "s"(g1) : "memory");
    __builtin_amdgcn_s_wait_tensorcnt(0);
  }

  const float* __restrict__ Xb   = Xm + (size_t)batch * KS * LL + nloc;
  const float* __restrict__ Arow = &As[lm * Kd];
  const int kph = 2 * half;

  v8f acc = {};
  v2f bcur;
  bcur[0] = Xb[(size_t)kph * LL];
  bcur[1] = Xb[(size_t)(kph + 1) * LL];

  for (int k0 = 0; k0 < Kd; k0 += 4) {
    v2f a = *(const v2f*)&Arow[k0 + kph];     // ds_load_b64 from staged tile
    v2f bnxt = {};
    if (k0 + 4 < Kd) {                        // uniform branch
      bnxt[0] = Xb[(size_t)(k0 + 4 + kph) * LL];
      bnxt[1] = Xb[(size_t)(k0 + 5 + kph) * LL];
    }
    if (k0 + 16 < Kd) {
      __builtin_prefetch(&Xb[(size_t)(k0 + 16) * LL], 0, 1);
    }
    acc = __builtin_amdgcn_wmma_f32_16x16x4_f32(
        /*neg_a=*/false, a, /*neg_b=*/false, bcur,
        /*c_mod=*/(short)0, acc, /*reuse_a=*/false, /*reuse_b=*/false);
    bcur = bnxt;
  }

  float* __restrict__ Yb = Ym + (size_t)batch * M * LL + nloc;
#pragma unroll
  for (int p = 0; p < 8; ++p) {
    const int row = m0 + p + 8 * half;
    if (row < M) {
      float v = acc[p];
      if (act == 1) {
        v += bias[row];
        v = (v > 20.0f) ? v : log1pf(__expf(v));   // softplus
      }
      Yb[(size_t)row * LL] = v;
    }
  }
}

// ---------------------------------------------------------------------------
// Depthwise 3x3 conv (pad 1) + bias + SiLU over [BT][DP][32][32]
// ---------------------------------------------------------------------------
__global__ void ss2d_dwconv_silu(const float* __restrict__ hin,
                                 const float* __restrict__ cw,
                                 const float* __restrict__ cb,
                                 float* __restrict__ hout) {
  int i = blockIdx.x * blockDim.x + threadIdx.x;
  if (i >= BT * DP * LL) return;
  int l  = i % LL;
  int d  = (i / LL) % DP;
  int hy = l >> 5, wx = l & 31;
  const float* __restrict__ base = hin + (size_t)(i - l);  // [bt][d] plane
  const float* __restrict__ wv = cw + d * 9;
  float s = 0.0f;
#pragma unroll
  for (int dy = -1; dy <= 1; ++dy) {
    int yy = hy + dy;
    if (yy < 0 || yy >= HH) continue;
#pragma unroll
    for (int dx = -1; dx <= 1; ++dx) {
      int xx = wx + dx;
      if (xx < 0 || xx >= WW) continue;
      s += wv[(dy + 1) * 3 + (dx + 1)] * base[yy * WW + xx];
    }
  }
  s += cb[d];
  hout[i] = s / (1.0f + __expf(-s));  // SiLU
}

// ---------------------------------------------------------------------------
// Scan-order -> natural (row-major) index within one 32x32 frame.
//  k=0: identity, k=1: transpose, k=2: reverse, k=3: transpose of reverse
// ---------------------------------------------------------------------------
__device__ __forceinline__ int ss2d_nat(int k, int j) {
  switch (k) {
    case 0:  return j;
    case 1:  return ((j & 31) << 5) | (j >> 5);
    case 2:  return (LL - 1) - j;
    default: { int jj = (LL - 1) - j; return ((jj & 31) << 5) | (jj >> 5); }
  }
}

// ---------------------------------------------------------------------------
// Selective scan. grid = (DP/128, KK, BB), block = 128 (one d per thread).
// State (N=16) and A-row live in registers. The shared B/C sequences are
// gathered straight into LDS with async loads (global_load_async_to_lds_b32,
// drained via s_wait_asynccnt) in 64-step chunks. Output is scattered to
// natural order so the 4 directions can simply be summed by the LN kernel.
// ---------------------------------------------------------------------------
#define SCH 64
__global__ void ss2d_selscan(const float* __restrict__ hbuf,   // [BT][DP][LL]
                             const float* __restrict__ xdbl,   // [KK][BT][RN][LL]
                             const float* __restrict__ delta,  // [KK][BT][DP][LL]
                             const float* __restrict__ A_logs, // [KK*DP][NN]
                             const float* __restrict__ Ds,     // [KK*DP]
                             float* __restrict__ ys)           // [KK][BT][DP][LL]
{
  const int d   = blockIdx.x * blockDim.x + threadIdx.x;
  const int k   = blockIdx.y;
  const int b   = blockIdx.z;
  const int tid = threadIdx.x;

  __shared__ float sBC[SCH * 32];  // 16 B + 16 C values per step

  float Arow[NN];
#pragma unroll
  for (int n = 0; n < NN; ++n)
    Arow[n] = -__expf(A_logs[(size_t)(k * DP + d) * NN + n]);
  const float Dd = Ds[k * DP + d];

  float st[NN];
#pragma unroll
  for (int n = 0; n < NN; ++n) st[n] = 0.0f;

  const unsigned sbase = (unsigned)(size_t)&sBC[0];

  for (int t = 0; t < TT; ++t) {
    const int bt = b * TT + t;
    const float* __restrict__ up = hbuf  + ((size_t)bt * DP + d) * LL;
    const float* __restrict__ dp = delta + (((size_t)k * BT + bt) * DP + d) * LL;
    const float* __restrict__ bc = xdbl  + (((size_t)k * BT + bt) * RN + RR) * LL;
    float* __restrict__ yp       = ys    + (((size_t)k * BT + bt) * DP + d) * LL;

    for (int jc = 0; jc < LL; jc += SCH) {
      __syncthreads();
      // Async gather of this chunk's B/C rows into LDS.
      for (int e = tid; e < SCH * 32; e += 128) {
        int s = e >> 5, r = e & 31;              // r<16 -> B row, r>=16 -> C row
        int l0 = ss2d_nat(k, jc + s);
        const float* src = bc + (size_t)r * LL + l0;
        unsigned dst = sbase + (unsigned)(e * 4);
        asm volatile("global_load_async_to_lds_b32 %0, %1, off"
                     :: "v"(dst), "v"(src) : "memory");
      }
      asm volatile("s_wait_asynccnt 0x0" ::: "memory");
      __syncthreads();

      for (int s = 0; s < SCH; ++s) {
        const int l0 = ss2d_nat(k, jc + s);
        const float u   = up[l0];
        const float dt  = dp[l0];
        const float dtu = dt * u;
        const float* __restrict__ Bv = &sBC[s * 32];
        const float* __restrict__ Cv = Bv + NN;
        float acc = 0.0f;
#pragma unroll
        for (int n = 0; n < NN; ++n) {
          float dA = __expf(dt * Arow[n]);
          st[n] = st[n] * dA + dtu * Bv[n];
          acc += st[n] * Cv[n];
        }
        yp[l0] = acc + Dd * u;   // scatter back to natural order
      }
    }
  }
}

// ---------------------------------------------------------------------------
// Sum 4 directions + LayerNorm over Dp.  grid = BT*LL blocks, block = 128.
// ---------------------------------------------------------------------------
__global__ void ss2d_merge_ln(const float* __restrict__ ys,  // [KK][BT][DP][LL]
                              const float* __restrict__ g,
                              const float* __restrict__ be,
                              float* __restrict__ yln) {     // [BT][DP][LL]
  const int pos = blockIdx.x;
  const int bt  = pos / LL;
  const int l   = pos % LL;
  const int tid = threadIdx.x;

  __shared__ float red[128];
  __shared__ float red2[128];

  float vals[3];
  float s = 0.0f, sq = 0.0f;
#pragma unroll
  for (int i = 0; i < 3; ++i) {
    int d = tid + i * 128;
    float v = 0.0f;
#pragma unroll
    for (int k = 0; k < KK; ++k)
      v += ys[(((size_t)k * BT + bt) * DP + d) * LL + l];
    vals[i] = v;
    s += v;
    sq += v * v;
  }
  red[tid] = s;
  red2[tid] = sq;
  __syncthreads();
  for (int off = 64; off > 0; off >>= 1) {
    if (tid < off) { red[tid] += red[tid + off]; red2[tid] += red2[tid + off]; }
    __syncthreads();
  }
  const float mu  = red[0] * (1.0f / DP);
  const float var = red2[0] * (1.0f / DP) - mu * mu;
  const float rs  = rsqrtf(var + 1e-5f);
#pragma unroll
  for (int i = 0; i < 3; ++i) {
    int d = tid + i * 128;
    yln[((size_t)bt * DP + d) * LL + l] = (vals[i] - mu) * rs * g[d] + be[d];
  }
}

// ---------------------------------------------------------------------------
// Scatter [bt][C][L] -> output (B,C,T,H,W)
// ---------------------------------------------------------------------------
__global__ void ss2d_scatter_out(const float* __restrict__ y5, float* __restrict__ out) {
  int i = blockIdx.x * blockDim.x + threadIdx.x;
  if (i >= BT * CC * LL) return;
  int l  = i % LL;
  int o  = (i / LL) % CC;
  int bt = i / (LL * CC);
  int b = bt / TT, t = bt % TT;
  out[(((size_t)(b * CC + o) * TT + t) * LL) + l] = y5[i];
}

// ---------------------------------------------------------------------------
extern "C" void kernel_launch(void* const* d_in, const int* in_sizes, int n_in,
                              void* d_out, int out_size, void* d_ws, size_t ws_size,
                              hipStream_t stream) {
  const float* x          = (const float*)d_in[0];
  const float* in_proj_w  = (const float*)d_in[1];
  const float* conv_w     = (const float*)d_in[2];
  const float* conv_b     = (const float*)d_in[3];
  const float* x_proj_w   = (const float*)d_in[4];
  const float* dt_w       = (const float*)d_in[5];
  const float* dt_b       = (const float*)d_in[6];
  const float* A_logs     = (const float*)d_in[7];
  const float* Ds         = (const float*)d_in[8];
  const float* ln_g       = (const float*)d_in[9];
  const float* ln_b       = (const float*)d_in[10];
  const float* out_proj_w = (const float*)d_in[11];
  float* out = (float*)d_out;
  float* ws  = (float*)d_ws;

  size_t off = 0;
  float* xf    = ws + off; off += (size_t)BT * CC * LL;        // 1.57M f
  float* hmid  = ws + off; off += (size_t)BT * DP * LL;        // 3.15M f
  float* hbuf  = ws + off; off += (size_t)BT * DP * LL;        // 3.15M f
  float* xdbl  = ws + off; off += (size_t)KK * BT * RN * LL;   // 1.44M f
  float* delta = ws + off; off += (size_t)KK * BT * DP * LL;   // 12.6M f
  float* ys    = ws + off; off += (size_t)KK * BT * DP * LL;   // 12.6M f
  float* yln = hmid;  // dead after conv -> reuse
  float* y5  = xf;    // dead after GEMM1 -> reuse

  const int NT = BT * LL / 16;  // 512 column tiles

  // 1) pack x -> xf
  {
    int tot = BT * CC * LL;
    ss2d_pack_xf<<<(tot + 255) / 256, 256, 0, stream>>>(x, xf);
  }
  // 2) in_proj:  hmid = in_proj_w (384x192) @ xf
  ss2d_gemm_wmma_f32<<<dim3(NT, DP / 16), 32, 0, stream>>>(
      in_proj_w, xf, hmid, nullptr, DP, CC, CC, 0);
  // 3) depthwise conv + SiLU
  {
    int tot = BT * DP * LL;
    ss2d_dwconv_silu<<<(tot + 255) / 256, 256, 0, stream>>>(hmid, conv_w, conv_b, hbuf);
  }
  // 4) x_proj per direction: xdbl[k] = x_proj_w[k] (44x384) @ h
  for (int k = 0; k < KK; ++k) {
    ss2d_gemm_wmma_f32<<<dim3(NT, (RN + 15) / 16), 32, 0, stream>>>(
        x_proj_w + (size_t)k * RN * DP, hbuf,
        xdbl + (size_t)k * BT * RN * LL, nullptr, RN, DP, DP, 0);
  }
  // 5) delta per direction: softplus(dt_w[k] (384x12) @ xdbl[k][0:12] + dt_b[k])
  for (int k = 0; k < KK; ++k) {
    ss2d_gemm_wmma_f32<<<dim3(NT, DP / 16), 32, 0, stream>>>(
        dt_w + (size_t)k * DP * RR, xdbl + (size_t)k * BT * RN * LL,
        delta + (size_t)k * BT * DP * LL, dt_b + (size_t)k * DP, DP, RR, RN, 1);
  }
  // 6) selective scan (4 directions, outputs scattered to natural order)
  ss2d_selscan<<<dim3(DP / 128, KK, BB), 128, 0, stream>>>(
      hbuf, xdbl, delta, A_logs, Ds, ys);
  // 7) merge 4 directions + LayerNorm over Dp
  ss2d_merge_ln<<<BT * LL, 128, 0, stream>>>(ys, ln_g, ln_b, yln);
  // 8) out_proj: y5 = out_proj_w (192x384) @ yln
  ss2d_gemm_wmma_f32<<<dim3(NT, CC / 16), 32, 0, stream>>>(
      out_proj_w, yln, y5, nullptr, CC, DP, DP, 0);
  // 9) scatter to (B,C,T,H,W)
  {
    int tot = BT * CC * LL;
    ss2d_scatter_out<<<(tot + 255) / 256, 256, 0, stream>>>(y5, out);
  }
}